// QuantumAttention_65481071410395
// MI455X (gfx1250) — compile-verified
//
#include <hip/hip_runtime.h>
#include <math.h>

// Problem constants (match reference)
#define B_   4
#define S_   4096
#define E_   1024
#define Q_   8
#define NTOK (B_ * S_)          // 16384 tokens
#define TILE_M 16               // tokens per wave tile
#define NTILES (NTOK / TILE_M)  // 1024
#define WAVES_PER_WG 8
#define THREADS (WAVES_PER_WG * 32)
#define NUM_WGS (NTILES / WAVES_PER_WG) // 128

typedef __attribute__((ext_vector_type(2))) float v2f;
typedef __attribute__((ext_vector_type(4))) float v4f;
typedef __attribute__((ext_vector_type(8))) float v8f;

// Fused: angles = x @ Wenc^T ; c = cos(angles); q = [prod(c[1:]), cumprod(c)[1:]];
// out = q @ Wdec^T + x.   All matrix math on V_WMMA_F32_16X16X4_F32 (wave32).
__global__ __launch_bounds__(THREADS)
void qattn_fused_kernel(const float* __restrict__ x,
                        const float* __restrict__ Wenc,   // [Q][E]
                        const float* __restrict__ Wdec,   // [E][Q]
                        float* __restrict__ out)
{
    // LDS: W_enc staged (hot: reread 256x per wave); per-wave transpose scratch.
    __shared__ float s_wenc[Q_ * E_];                       // 32 KB
    __shared__ float s_ang[WAVES_PER_WG][TILE_M * 17];      // 16x16 tile, stride 17 (bank-conflict pad)
    __shared__ float s_qv [WAVES_PER_WG][TILE_M * Q_];      // 16x8 q-values

    const int tid = threadIdx.x;

    // ---- Stage W_enc into LDS (wide b128 copies) ----
    {
        const v4f* src = (const v4f*)Wenc;
        v4f*       dst = (v4f*)s_wenc;
        const int  n4  = (Q_ * E_) / 4;     // 2048
        for (int i = tid; i < n4; i += THREADS) dst[i] = src[i];
    }
    __syncthreads();

    const int wave = tid >> 5;
    const int lane = tid & 31;
    const int m16  = lane & 15;            // M row (A) / N col (B,C,D) index
    const int kk   = (lane >> 4) << 1;     // K sub-pair: 0 (lanes 0-15) or 2 (lanes 16-31)
    const int mhalf = (lane >> 4) << 3;    // C/D row base: 0 or 8

    const int tile = blockIdx.x * WAVES_PER_WG + wave;     // 0..NTILES-1
    const long tok0 = (long)tile * TILE_M;
    const float* xt = x + tok0 * E_;
    float*       ot = out + tok0 * E_;

    // =========== Phase 1: encode (16xE) x (Ex16) -> angles 16x16 (cols 0..7 valid) ===========
    // A frag (16x4 f32): lane holds x[m16][k+kk], x[m16][k+kk+1]
    // B frag (4x16 f32): lane holds Wenc[n][k+kk], Wenc[n][k+kk+1]  (n>=8 -> dup row, col ignored)
    v8f acc = {};
    {
        const float* arow = xt + (long)m16 * E_ + kk;
        const float* brow = s_wenc + (m16 & 7) * E_ + kk;
        #pragma unroll 8
        for (int k = 0; k < E_; k += 4) {
            v2f a = *(const v2f*)(arow + k);
            v2f b = *(const v2f*)(brow + k);
            acc = __builtin_amdgcn_wmma_f32_16x16x4_f32(
                false, a, false, b, (short)0, acc, false, false);
        }
    }

    // =========== Phase 2: cos + prefix products (wave-local LDS transpose) ===========
    // acc[i]: lanes 0-15 -> angles[token=i][q=lane], lanes 16-31 -> angles[token=i+8][q=lane-16]
    {
        float* ap = &s_ang[wave][0];
        #pragma unroll
        for (int i = 0; i < 8; ++i)
            ap[(mhalf + i) * 17 + m16] = acc[i];
    }
    __builtin_amdgcn_wave_barrier();   // same-wave DS ops are in-order; fence compiler

    if (lane < 16) {
        const float* ap = &s_ang[wave][lane * 17];
        float c[Q_];
        #pragma unroll
        for (int q = 0; q < Q_; ++q) c[q] = cosf(ap[q]);
        float p[Q_];
        p[0] = c[0];
        #pragma unroll
        for (int q = 1; q < Q_; ++q) p[q] = p[q - 1] * c[q];
        float q0 = c[1];
        #pragma unroll
        for (int q = 2; q < Q_; ++q) q0 *= c[q];
        float* qp = &s_qv[wave][lane * Q_];
        qp[0] = q0;
        #pragma unroll
        for (int q = 1; q < Q_; ++q) qp[q] = p[q];
    }
    __builtin_amdgcn_wave_barrier();

    // =========== Phase 3: decode (16x8) x (8xE) + residual ===========
    // A frags (two K=4 slices of the 16x8 q-matrix), read once from LDS.
    v2f a0, a1;
    {
        const float* qp = &s_qv[wave][m16 * Q_];
        a0 = *(const v2f*)(qp + kk);        // K = kk, kk+1
        a1 = *(const v2f*)(qp + 4 + kk);    // K = 4+kk, 5+kk
    }

    #pragma unroll 2
    for (int nb = 0; nb < E_; nb += 16) {
        // B frags: B[k][n] = Wdec[nb+n][k]; Wdec row stride = Q_ = 8 floats (32B resident in L2)
        const float* wd = Wdec + (nb + m16) * Q_;
        v2f b0 = *(const v2f*)(wd + kk);
        v2f b1 = *(const v2f*)(wd + 4 + kk);
        v8f d = {};
        d = __builtin_amdgcn_wmma_f32_16x16x4_f32(false, a0, false, b0, (short)0, d, false, false);
        d = __builtin_amdgcn_wmma_f32_16x16x4_f32(false, a1, false, b1, (short)0, d, false, false);
        // residual + store: D vgpr i -> token (mhalf+i), column e = nb + m16
        #pragma unroll
        for (int i = 0; i < 8; ++i) {
            const int idx = (mhalf + i) * E_ + nb + m16;
            ot[idx] = d[i] + xt[idx];       // x re-read hits L2 (x fits in 192MB L2)
        }
    }
}

extern "C" void kernel_launch(void* const* d_in, const int* in_sizes, int n_in,
                              void* d_out, int out_size, void* d_ws, size_t ws_size,
                              hipStream_t stream) {
    const float* x    = (const float*)d_in[0];   // [B,S,E] fp32
    const float* Wenc = (const float*)d_in[1];   // [Q,E]   fp32
    const float* Wdec = (const float*)d_in[2];   // [E,Q]   fp32
    float* out        = (float*)d_out;           // [B,S,E] fp32
    (void)in_sizes; (void)n_in; (void)out_size; (void)d_ws; (void)ws_size;

    qattn_fused_kernel<<<NUM_WGS, THREADS, 0, stream>>>(x, Wenc, Wdec, out);
}